// attention_pb_32547262169742
// MI455X (gfx1250) — compile-verified
//
#include <hip/hip_runtime.h>
#include <hip/hip_bf16.h>

typedef __attribute__((ext_vector_type(2))) float v2f;
typedef __attribute__((ext_vector_type(8))) float v8f;

#define C_DIM   96
#define T_DIM   1024
#define V_DIM   10
#define TV      10240        // T*V
#define S_DIM   3
#define MID     32
#define QK_ROWS 192          // q(96) + k(96) rows of W_qkv
#define TBLK    16           // t-blocks per n
#define CHUNK   160          // columns (t,v) per inner chunk = 16 t's
#define NCHUNK  4
#define COLS_PER_BLK (CHUNK*NCHUNK)  // 640 = 64 t's
#define NTHREADS 256

__constant__ int c_PB[10] = {4, 4, 0, 0, 1, 1, 2, 2, 3, 3};

// ---------------------------------------------------------------------------
// Kernel 1: Q/K GEMM via WMMA f32 16x16x4 + per-block Gram partial sums.
// grid = N * TBLK blocks, 256 threads (8 waves).
// LDS: sW[192*96] | sXT[160*96] | sQK[192*160] | sB[192]
// ---------------------------------------------------------------------------
__global__ __launch_bounds__(NTHREADS) void k_qk_gram(
    const float* __restrict__ x, const float* __restrict__ Wqkv,
    const float* __restrict__ bqkv, float* __restrict__ gpart)
{
    extern __shared__ float smem[];
    float* sW  = smem;                       // 18432
    float* sXT = sW + QK_ROWS * C_DIM;       // 15360  [col][c]
    float* sQK = sXT + CHUNK * C_DIM;        // 30720  [o][col]
    float* sB  = sQK + QK_ROWS * CHUNK;      // 192

    const int tid    = threadIdx.x;
    const int lane   = tid & 31;
    const int wave   = tid >> 5;
    const int laneM  = lane & 15;
    const int laneHi = lane >> 4;
    const int n      = blockIdx.x / TBLK;
    const int tblk   = blockIdx.x % TBLK;
    const int pBase0 = tblk * COLS_PER_BLK;

    for (int i = tid; i < QK_ROWS * C_DIM; i += NTHREADS) sW[i] = Wqkv[i];
    for (int i = tid; i < QK_ROWS; i += NTHREADS)         sB[i] = bqkv[i];

    float g0 = 0.f, g1 = 0.f;  // up to 2 gram combos per thread
    const float* xn = x + (size_t)n * C_DIM * TV;

    for (int ch = 0; ch < NCHUNK; ++ch) {
        const int pBase = pBase0 + ch * CHUNK;
        __syncthreads();
        // load x chunk transposed: sXT[col*96 + c]
        for (int i = tid; i < CHUNK * C_DIM; i += NTHREADS) {
            int c = i / CHUNK, j = i % CHUNK;
            sXT[j * C_DIM + c] = xn[(size_t)c * TV + pBase + j];
        }
        __syncthreads();

        // 12 o-tiles x 10 col-tiles of 16x16, K accumulated 96/4 = 24 WMMAs
        for (int tile = wave; tile < 120; tile += 8) {
            const int o0   = (tile / 10) * 16;
            const int col0 = (tile % 10) * 16;
            v8f acc = {};
            const float* aRow = sW  + (o0   + laneM) * C_DIM + 2 * laneHi;
            const float* bRow = sXT + (col0 + laneM) * C_DIM + 2 * laneHi;
#pragma unroll
            for (int k0 = 0; k0 < C_DIM; k0 += 4) {
                v2f a = *(const v2f*)(aRow + k0);
                v2f b = *(const v2f*)(bRow + k0);
                acc = __builtin_amdgcn_wmma_f32_16x16x4_f32(
                    false, a, false, b, (short)0, acc, false, false);
            }
#pragma unroll
            for (int r = 0; r < 8; ++r) {
                int o = o0 + r + 8 * laneHi;
                sQK[o * CHUNK + col0 + laneM] = acc[r] + sB[o];
            }
        }
        __syncthreads();

        // Gram: G_s[u][w] += sum_{c,t} q[s*32+c, t*10+u] * k[96+s*32+c, t*10+w]
        int ci = 0;
        for (int combo = tid; combo < 300; combo += NTHREADS, ++ci) {
            const int s = combo / 100;
            const int u = (combo / 10) % 10;
            const int w = combo % 10;
            const float* qr = sQK + (s * MID) * CHUNK + u;
            const float* kr = sQK + (96 + s * MID) * CHUNK + w;
            float sum = 0.f;
            for (int c = 0; c < MID; ++c) {
                const float* q = qr + c * CHUNK;
                const float* k = kr + c * CHUNK;
#pragma unroll
                for (int tt = 0; tt < 16; ++tt)
                    sum += q[tt * 10] * k[tt * 10];
            }
            if (ci == 0) g0 += sum; else g1 += sum;
        }
    }

    int ci = 0;
    for (int combo = tid; combo < 300; combo += NTHREADS, ++ci) {
        float g = (ci == 0) ? g0 : g1;
        gpart[(size_t)(n * TBLK + tblk) * 300 + combo] = g;
    }
}

// ---------------------------------------------------------------------------
// Kernel 2: reduce Gram partials -> attention0 (to output tail) and
// attention = attention0 + b_att[PART_BODY gather] (to workspace).
// ---------------------------------------------------------------------------
__global__ __launch_bounds__(NTHREADS) void k_att(
    const float* __restrict__ gpart, const float* __restrict__ b_att,
    float* __restrict__ att0_out, float* __restrict__ att_ws)
{
    int idx = blockIdx.x * NTHREADS + threadIdx.x;
    if (idx >= 32 * 300) return;
    const int n = idx / 300;
    const int combo = idx % 300;
    const int s = combo / 100, u = (combo / 10) % 10, w = combo % 10;
    float sum = 0.f;
    for (int tb = 0; tb < TBLK; ++tb)
        sum += gpart[(size_t)(n * TBLK + tb) * 300 + combo];
    float a0 = tanhf(sum * (1.0f / 32768.0f));
    att0_out[idx] = a0;
    float bb = b_att[n * 75 + s * 25 + c_PB[u] * 5 + c_PB[w]];
    att_ws[idx] = a0 + bb;
}

// ---------------------------------------------------------------------------
// Kernel 3: V GEMM (WMMA) -> attention apply (VALU) -> FF GEMM (WMMA) ->
// bias + BN + residual + leaky ReLU epilogue.
// LDS: sWv[96*96] | sWff[96*96] | sXT[160*96] | sV[96*160] | sY1T[160*96]
//      | sAtt[300] | sBv[96] sBff[96] sScale[96] sShift[96]
// ---------------------------------------------------------------------------
__global__ __launch_bounds__(NTHREADS) void k_out(
    const float* __restrict__ x, const float* __restrict__ Wqkv,
    const float* __restrict__ bqkv, const float* __restrict__ Wff,
    const float* __restrict__ bff, const float* __restrict__ gamma,
    const float* __restrict__ beta, const float* __restrict__ mean,
    const float* __restrict__ var, const float* __restrict__ att_ws,
    float* __restrict__ out)
{
    extern __shared__ float smem[];
    float* sWv    = smem;                 // 9216
    float* sWff   = sWv + 9216;           // 9216
    float* sXT    = sWff + 9216;          // 15360 [col][c]
    float* sV     = sXT + CHUNK * C_DIM;  // 15360 [c][col]
    float* sY1T   = sV + C_DIM * CHUNK;   // 15360 [col][c]
    float* sAtt   = sY1T + CHUNK * C_DIM; // 300
    float* sBv    = sAtt + 300;           // 96
    float* sBff   = sBv + 96;             // 96
    float* sScale = sBff + 96;            // 96
    float* sShift = sScale + 96;          // 96

    const int tid    = threadIdx.x;
    const int lane   = tid & 31;
    const int wave   = tid >> 5;
    const int laneM  = lane & 15;
    const int laneHi = lane >> 4;
    const int n      = blockIdx.x / TBLK;
    const int tblk   = blockIdx.x % TBLK;
    const int pBase0 = tblk * COLS_PER_BLK;

    for (int i = tid; i < C_DIM * C_DIM; i += NTHREADS) {
        sWv[i]  = Wqkv[QK_ROWS * C_DIM + i];  // W_v rows 192..287
        sWff[i] = Wff[i];
    }
    for (int i = tid; i < 300; i += NTHREADS) sAtt[i] = att_ws[n * 300 + i];
    if (tid < C_DIM) {
        sBv[tid]  = bqkv[QK_ROWS + tid];
        sBff[tid] = bff[tid];
        float sc = gamma[tid] * rsqrtf(var[tid] + 1e-5f);
        sScale[tid] = sc;
        sShift[tid] = beta[tid] - mean[tid] * sc;
    }

    const float* xn = x + (size_t)n * C_DIM * TV;
    float* outn = out + (size_t)n * C_DIM * TV;

    for (int ch = 0; ch < NCHUNK; ++ch) {
        const int pBase = pBase0 + ch * CHUNK;
        __syncthreads();
        for (int i = tid; i < CHUNK * C_DIM; i += NTHREADS) {
            int c = i / CHUNK, j = i % CHUNK;
            sXT[j * C_DIM + c] = xn[(size_t)c * TV + pBase + j];
        }
        __syncthreads();

        // GEMM1: V = Wv * X  (6 x 10 tiles)
        for (int tile = wave; tile < 60; tile += 8) {
            const int o0   = (tile / 10) * 16;
            const int col0 = (tile % 10) * 16;
            v8f acc = {};
            const float* aRow = sWv + (o0   + laneM) * C_DIM + 2 * laneHi;
            const float* bRow = sXT + (col0 + laneM) * C_DIM + 2 * laneHi;
#pragma unroll
            for (int k0 = 0; k0 < C_DIM; k0 += 4) {
                v2f a = *(const v2f*)(aRow + k0);
                v2f b = *(const v2f*)(bRow + k0);
                acc = __builtin_amdgcn_wmma_f32_16x16x4_f32(
                    false, a, false, b, (short)0, acc, false, false);
            }
#pragma unroll
            for (int r = 0; r < 8; ++r)
                sV[(o0 + r + 8 * laneHi) * CHUNK + col0 + laneM] = acc[r];
        }
        __syncthreads();

        // attention apply: y1[c, t*10+u] = sum_w att[s(c),u,w]*(v[c,t*10+w]+bv[c])
        for (int e = tid; e < C_DIM * CHUNK; e += NTHREADS) {
            const int c   = e / CHUNK;
            const int col = e % CHUNK;
            const int u   = col % 10;
            const int t10 = col - u;
            const int s   = c >> 5;
            const float bv = sBv[c];
            const float* ar = sAtt + s * 100 + u * 10;
            const float* vr = sV + c * CHUNK + t10;
            float y = 0.f;
#pragma unroll
            for (int w = 0; w < 10; ++w) y += ar[w] * (vr[w] + bv);
            sY1T[col * C_DIM + c] = y;
        }
        __syncthreads();

        // GEMM2: Y2 = Wff * Y1, fused epilogue
        for (int tile = wave; tile < 60; tile += 8) {
            const int o0   = (tile / 10) * 16;
            const int col0 = (tile % 10) * 16;
            v8f acc = {};
            const float* aRow = sWff + (o0   + laneM) * C_DIM + 2 * laneHi;
            const float* bRow = sY1T + (col0 + laneM) * C_DIM + 2 * laneHi;
#pragma unroll
            for (int k0 = 0; k0 < C_DIM; k0 += 4) {
                v2f a = *(const v2f*)(aRow + k0);
                v2f b = *(const v2f*)(bRow + k0);
                acc = __builtin_amdgcn_wmma_f32_16x16x4_f32(
                    false, a, false, b, (short)0, acc, false, false);
            }
#pragma unroll
            for (int r = 0; r < 8; ++r) {
                const int o   = o0 + r + 8 * laneHi;
                const int col = col0 + laneM;
                float y2 = (acc[r] + sBff[o]) * sScale[o] + sShift[o];
                float z  = sXT[col * C_DIM + o] + y2;
                outn[(size_t)o * TV + pBase + col] = (z > 0.f) ? z : 0.1f * z;
            }
        }
    }
}

// ---------------------------------------------------------------------------
extern "C" void kernel_launch(void* const* d_in, const int* in_sizes, int n_in,
                              void* d_out, int out_size, void* d_ws, size_t ws_size,
                              hipStream_t stream) {
    const float* x     = (const float*)d_in[0];
    const float* b_att = (const float*)d_in[1];
    const float* Wqkv  = (const float*)d_in[2];
    const float* bqkv  = (const float*)d_in[3];
    const float* Wff   = (const float*)d_in[4];
    const float* bff   = (const float*)d_in[5];
    const float* gamma = (const float*)d_in[6];
    const float* beta  = (const float*)d_in[7];
    const float* mean  = (const float*)d_in[8];
    const float* var   = (const float*)d_in[9];
    float* out = (float*)d_out;

    float* gpart  = (float*)d_ws;              // 32*16*300 = 153600 floats
    float* att_ws = gpart + 32 * TBLK * 300;   // 9600 floats

    const size_t lds1 = (size_t)(QK_ROWS * C_DIM + CHUNK * C_DIM +
                                 QK_ROWS * CHUNK + QK_ROWS) * sizeof(float);
    k_qk_gram<<<32 * TBLK, NTHREADS, lds1, stream>>>(x, Wqkv, bqkv, gpart);

    k_att<<<(32 * 300 + NTHREADS - 1) / NTHREADS, NTHREADS, 0, stream>>>(
        gpart, b_att, out + (size_t)32 * C_DIM * TV, att_ws);

    const size_t lds3 = (size_t)(2 * 9216 + 3 * CHUNK * C_DIM + 300 + 4 * 96) *
                        sizeof(float);
    k_out<<<32 * TBLK, NTHREADS, lds3, stream>>>(x, Wqkv, bqkv, Wff, bff, gamma,
                                                 beta, mean, var, att_ws, out);
}